// FilterDetection_15375982920328
// MI455X (gfx1250) — compile-verified
//
#include <hip/hip_runtime.h>
#include <cstdint>

#define DEV __device__ __forceinline__

constexpr int   Bc   = 4;
constexpr int   Nc   = 100000;
constexpr int   Cc   = 80;
constexpr int   NCc  = Nc * Cc;        // 8,000,000 per image
constexpr int   Kc   = 4096;           // PERFORMANCE_COUNT
constexpr int   Pc   = 100;            // PROPOSAL_COUNT
constexpr float SCORE_TH = 0.05f;
constexpr float IOU_TH   = 0.5f;

// ---- workspace layout (uint32 units) ----
constexpr int WS_H1    = 0;                     // B*2048 radix pass1 hist (float bits 30..20)
constexpr int WS_H2    = WS_H1   + Bc * 2048;   // B*2048 pass2 (bits 19..9)
constexpr int WS_H3    = WS_H2   + Bc * 2048;   // B*512  pass3 (bits 8..0)
constexpr int WS_INFO  = WS_H3   + Bc * 512;    // B*8 select state
constexpr int WS_CNT   = WS_INFO + Bc * 8;      // B*4 compaction counters
constexpr int WS_SCORE = WS_CNT  + Bc * 4;      // B*K f32 candidate scores
constexpr int WS_IDX   = WS_SCORE+ Bc * Kc;     // B*K i32 flat n*C+c
constexpr int WS_BOX   = WS_IDX  + Bc * Kc;     // B*K*4 f32 decoded boxes (16B aligned)
constexpr int WS_CLS   = WS_BOX  + Bc * Kc * 4; // B*K i32
constexpr int WS_ANCH  = WS_CLS  + Bc * Kc;     // B*K i32
constexpr int WS_SEL   = WS_ANCH + Bc * Kc;     // B*P i32 NMS picks
constexpr int WS_END   = WS_SEL  + Bc * Pc;
constexpr size_t MASK_BYTE = (((size_t)WS_END * 4 + 255) / 256) * 256; // 8MB bitmask region

typedef __attribute__((ext_vector_type(2))) float v2f;
typedef __attribute__((ext_vector_type(8))) float v8f;

DEV float fused_score(const float* __restrict__ logits, const float* __restrict__ ctr,
                      int b, int e) {
  int n = e / Cc;
  float l  = logits[(size_t)b * NCc + e];
  float ce = ctr[(size_t)b * Nc + n];
  return sqrtf(l * ce);
}

// ---------------- init ----------------
__global__ void k_init(uint32_t* __restrict__ ws) {
  int i = blockIdx.x * 256 + threadIdx.x;
  const int HT = Bc * (2048 + 2048 + 512);
  if (i < HT) ws[i] = 0u;
  if (i < Bc * Kc) {
    ((float*)(ws + WS_SCORE))[i] = 0.0f;
    ((int*)(ws + WS_IDX))[i]     = 0x7FFFFFFF;
  }
}

// ---------------- radix histogram (3 passes on float bit-fields) ----------------
__global__ void k_hist(const float* __restrict__ logits, const float* __restrict__ ctr,
                       uint32_t* __restrict__ ws, int pass) {
  __shared__ uint32_t h[2048];
  const int b = blockIdx.y, tid = threadIdx.x;
  for (int i = tid; i < 2048; i += 256) h[i] = 0u;
  __syncthreads();
  uint32_t t1 = 0, t2 = 0;
  if (pass >= 2) t1 = ws[WS_INFO + b * 8 + 0];
  if (pass == 3) t2 = ws[WS_INFO + b * 8 + 2];
  const int base = blockIdx.x * (256 * 8);
  if (base + 16384 + tid < NCc)   // gfx1250 global_prefetch_b8 path
    __builtin_prefetch(&logits[(size_t)b * NCc + base + 16384 + tid], 0, 1);
  for (int k = 0; k < 8; ++k) {
    int e = base + k * 256 + tid;
    if (e >= NCc) break;
    int c = e % Cc;
    if (c == 0) continue;                       // IGNORE_LABEL
    float s = fused_score(logits, ctr, b, e);
    if (!(s > SCORE_TH)) continue;
    uint32_t bits = __float_as_uint(s);         // positive floats: bit order == value order
    if (pass == 1)       atomicAdd(&h[bits >> 20], 1u);
    else if (pass == 2) { if ((bits >> 20) == t1) atomicAdd(&h[(bits >> 9) & 0x7FFu], 1u); }
    else                { if ((bits >> 9) == ((t1 << 11) | t2)) atomicAdd(&h[bits & 0x1FFu], 1u); }
  }
  __syncthreads();
  uint32_t* g = ws + (pass == 1 ? WS_H1 + b * 2048
                     : pass == 2 ? WS_H2 + b * 2048
                                 : WS_H3 + b * 512);
  const int nb = (pass == 3) ? 512 : 2048;
  for (int i = tid; i < nb; i += 256) if (h[i]) atomicAdd(&g[i], h[i]);
}

// ---------------- radix select (tiny, one thread per image) ----------------
__global__ void k_select(uint32_t* __restrict__ ws, int pass) {
  if (threadIdx.x != 0) return;
  const int b = blockIdx.x;
  uint32_t* info = ws + WS_INFO + b * 8;
  if (pass == 1) {
    const uint32_t* h = ws + WS_H1 + b * 2048;
    uint32_t cum = 0; int t = 0; uint32_t above = 0; bool f = false;
    for (int i = 2047; i >= 0; --i) {
      if (cum + h[i] >= (uint32_t)Kc) { t = i; above = cum; f = true; break; }
      cum += h[i];
    }
    if (!f) { t = 0; above = cum; }
    info[0] = (uint32_t)t; info[1] = above;
  } else if (pass == 2) {
    const uint32_t* h = ws + WS_H2 + b * 2048;
    uint32_t target = (uint32_t)Kc - info[1];
    uint32_t cum = 0; int t = 0;
    for (int i = 2047; i >= 0; --i) {
      if (cum + h[i] >= target) { t = i; break; }
      cum += h[i];
    }
    info[2] = (uint32_t)t; info[3] = info[1] + cum;
  } else {
    const uint32_t* h = ws + WS_H3 + b * 512;
    uint32_t target = (uint32_t)Kc - info[3];
    uint32_t cum = 0; int t = 0;
    for (int i = 511; i >= 0; --i) {
      if (cum + h[i] >= target) { t = i; break; }
      cum += h[i];
    }
    uint32_t cg = info[3] + cum;                // count strictly greater than threshold
    info[4] = (uint32_t)t; info[5] = cg;
    info[6] = (info[0] << 20) | (info[2] << 9) | (uint32_t)t;  // exact threshold bits
    ws[WS_CNT + b * 4 + 0] = 0u;   // greater-than counter
    ws[WS_CNT + b * 4 + 1] = cg;   // equal counter starts after the strict-greater block
  }
}

// ---------------- compaction of the exact top-K set ----------------
__global__ void k_compact(const float* __restrict__ logits, const float* __restrict__ ctr,
                          uint32_t* __restrict__ ws) {
  const int b = blockIdx.y, tid = threadIdx.x;
  const uint32_t vthr = ws[WS_INFO + b * 8 + 6];
  float* cs = (float*)(ws + WS_SCORE) + (size_t)b * Kc;
  int*   ci = (int*)(ws + WS_IDX) + (size_t)b * Kc;
  uint32_t* cnt = ws + WS_CNT + b * 4;
  const int base = blockIdx.x * (256 * 8);
  for (int k = 0; k < 8; ++k) {
    int e = base + k * 256 + tid;
    if (e >= NCc) break;
    int c = e % Cc;
    if (c == 0) continue;
    float s = fused_score(logits, ctr, b, e);
    if (!(s > SCORE_TH)) continue;
    uint32_t bits = __float_as_uint(s);
    if (bits < vthr) continue;
    uint32_t pos;
    if (bits > vthr) pos = atomicAdd(&cnt[0], 1u);            // always < K by construction
    else { pos = atomicAdd(&cnt[1], 1u); if (pos >= (uint32_t)Kc) continue; }
    cs[pos] = s; ci[pos] = e;
  }
}

// ---------------- bitonic sort (desc score, asc index ties) + box decode ----------------
DEV bool rank_before(float sa, int ia, float sb, int ib) {
  return (sa > sb) || (sa == sb && ia < ib);
}

__global__ __launch_bounds__(1024) void k_sort_decode(const float* __restrict__ regress,
                                                      const float* __restrict__ points,
                                                      uint32_t* __restrict__ ws) {
  __shared__ float ss[Kc];
  __shared__ int   si[Kc];
  const int b = blockIdx.x, tid = threadIdx.x;
  float* cs = (float*)(ws + WS_SCORE) + (size_t)b * Kc;
  int*   ci = (int*)(ws + WS_IDX) + (size_t)b * Kc;
  for (int i = tid; i < Kc; i += 1024) { ss[i] = cs[i]; si[i] = ci[i]; }
  __syncthreads();
  for (int k = 2; k <= Kc; k <<= 1) {
    for (int j = k >> 1; j > 0; j >>= 1) {
      for (int i = tid; i < Kc; i += 1024) {
        int l = i ^ j;
        if (l > i) {
          bool desc = ((i & k) == 0);
          bool sw = desc ? rank_before(ss[l], si[l], ss[i], si[i])
                         : rank_before(ss[i], si[i], ss[l], si[l]);
          if (sw) {
            float ts = ss[i]; ss[i] = ss[l]; ss[l] = ts;
            int   ti = si[i]; si[i] = si[l]; si[l] = ti;
          }
        }
      }
      __syncthreads();
    }
  }
  float* cb = (float*)(ws + WS_BOX) + (size_t)b * Kc * 4;
  int*   cc = (int*)(ws + WS_CLS) + (size_t)b * Kc;
  int*   ca = (int*)(ws + WS_ANCH) + (size_t)b * Kc;
  for (int i = tid; i < Kc; i += 1024) {
    float s = ss[i]; int idx = si[i];
    cs[i] = s; ci[i] = idx;
    if (s > 0.0f && idx < NCc) {
      int anchor = idx / Cc, cls = idx % Cc;
      float px = points[anchor * 2 + 0], py = points[anchor * 2 + 1];
      const float* rg = regress + ((size_t)b * Nc + anchor) * 4;
      cb[i * 4 + 0] = fminf(fmaxf(px - rg[0], 0.f), 1.f);
      cb[i * 4 + 1] = fminf(fmaxf(py - rg[1], 0.f), 1.f);
      cb[i * 4 + 2] = fminf(fmaxf(px + rg[2], 0.f), 1.f);
      cb[i * 4 + 3] = fminf(fmaxf(py + rg[3], 0.f), 1.f);
      cc[i] = cls; ca[i] = anchor;
    } else {
      cb[i * 4 + 0] = 0.f; cb[i * 4 + 1] = 0.f; cb[i * 4 + 2] = 0.f; cb[i * 4 + 3] = 0.f;
      cc[i] = -1; ca[i] = 0;
    }
  }
}

// ---------------- IoU suppression bitmask: WMMA pairwise area-sum + VALU intersection ----
// One wave per 16x16 tile. A(16x4 f32)=[area_i,1,0,0], B(4x16)=[1;area_j;0;0] ->
// C[i][j] = area_i + area_j on the matrix engine; C/D lane layout (VGPR v: M=v lanes 0-15,
// M=v+8 lanes 16-31, N=lane&15) matches ballot bit packing exactly.
// Row data per step is half-wave-uniform, so it is re-read directly from the hot 64KB/image
// candidate arrays (L0/L2 resident) instead of ds_bpermute broadcasts; the divide-free
// predicate inter > 0.5*max(union,1e-9) replaces the IEEE division (0.5*x is exact).
__global__ void k_iou_mask(const uint32_t* __restrict__ ws,
                           unsigned short* __restrict__ mask) {
  const int b    = blockIdx.y;
  const int lane = threadIdx.x & 31;
  const int wave = threadIdx.x >> 5;
  const int tile = blockIdx.x * 8 + wave;     // 256x256 tiles per image
  const int ty = tile >> 8, tx = tile & 255;
  const int li = lane & 15;
  const float4* box = (const float4*)((const float*)(ws + WS_BOX) + (size_t)b * Kc * 4);
  const int*    cls = (const int*)(ws + WS_CLS) + (size_t)b * Kc;
  const int rI = ty * 16 + li, cI = tx * 16 + li;
  const float4 cbx = box[cI];                 // this lane's column box
  const int    ccl = cls[cI];
  const float  car = (cbx.z - cbx.x) * (cbx.w - cbx.y);
  const float4 rbx = box[rI];                 // row box li (for the WMMA A fragment)
  const float  rar = (rbx.z - rbx.x) * (rbx.w - rbx.y);
  const bool lo = lane < 16;
  // A 16x4 f32 layout: lanes 0-15 VGPR0=K0,VGPR1=K1; lanes 16-31 VGPR0=K2,VGPR1=K3 (zeros)
  v2f A;  A[0]  = lo ? rar : 0.f;  A[1]  = lo ? 1.f : 0.f;
  v2f Bm; Bm[0] = lo ? 1.f : 0.f;  Bm[1] = lo ? car : 0.f;   // B rows K0=ones, K1=area_j
  v8f Cz = {};
  v8f S = __builtin_amdgcn_wmma_f32_16x16x4_f32(false, A, false, Bm, (short)0, Cz, false, false);
  unsigned short* mrow = mask + ((size_t)b * Kc + (size_t)ty * 16) * 256 + tx;
  const int rbase = ty * 16 + (lo ? 0 : 8);   // rows owned by this half-wave
#pragma unroll
  for (int v = 0; v < 8; ++v) {
    const float4 rb = box[rbase + v];         // half-wave-uniform, cache-hot
    const int   bcl = cls[rbase + v];
    float ltx = fmaxf(rb.x, cbx.x), lty = fmaxf(rb.y, cbx.y);
    float rbr = fminf(rb.z, cbx.z), rbb = fminf(rb.w, cbx.w);
    float w = fmaxf(rbr - ltx, 0.f), hgt = fmaxf(rbb - lty, 0.f);
    float inter = w * hgt;
    float uni = S[v] - inter;                 // area_i + area_j from WMMA
    bool sup = (inter > IOU_TH * fmaxf(uni, 1e-9f)) && (bcl == ccl);
    unsigned int bal = __builtin_amdgcn_ballot_w32(sup); // bits 15:0 row v, 31:16 row v+8
    if (lane == 0) {
      mrow[(size_t)v * 256]       = (unsigned short)(bal & 0xFFFFu);
      mrow[(size_t)(v + 8) * 256] = (unsigned short)(bal >> 16);
    }
  }
}

// ---------------- greedy NMS: sorted order => first set bit == argmax ----------------
__global__ void k_nms(uint32_t* __restrict__ ws, const unsigned short* __restrict__ mask) {
  __shared__ uint32_t valid[128];
  __shared__ int firstw;
  const int b = blockIdx.x, tid = threadIdx.x;
  const float* cs = (const float*)(ws + WS_SCORE) + (size_t)b * Kc;
  const unsigned short* mb = mask + (size_t)b * Kc * 256;
  int* sel = (int*)(ws + WS_SEL) + b * Pc;
  if (tid < 128) {
    uint32_t v = 0;
    for (int t = 0; t < 32; ++t)
      if (cs[tid * 32 + t] > 0.f) v |= (1u << t);
    valid[tid] = v;
  }
  __syncthreads();
  for (int p = 0; p < Pc; ++p) {
    if (tid == 0) firstw = 128;
    __syncthreads();
    if (tid < 128 && valid[tid] != 0u) atomicMin(&firstw, tid);
    __syncthreads();
    int j = -1;
    if (firstw < 128) j = firstw * 32 + __ffs(valid[firstw]) - 1;
    if (tid == 0) sel[p] = j;
    if (j >= 0 && tid < 128) {
      const unsigned short* row = mb + (size_t)j * 256;
      uint32_t m = (uint32_t)row[2 * tid] | ((uint32_t)row[2 * tid + 1] << 16);
      uint32_t nv = valid[tid] & ~m;
      if (tid == (j >> 5)) nv &= ~(1u << (j & 31));  // diagonal clears j anyway; be explicit
      valid[tid] = nv;
    }
    __syncthreads();
  }
}

// ---------------- output gather ----------------
__global__ void k_gather(const float* __restrict__ logits, const float* __restrict__ ctr,
                         const uint32_t* __restrict__ ws, float* __restrict__ out) {
  const int b = blockIdx.y, p = blockIdx.x, tid = threadIdx.x;
  const int* sel = (const int*)(ws + WS_SEL) + b * Pc;
  int j = sel[p];
  bool ok = j >= 0;
  int jc = ok ? j : 0;
  const int*   ca = (const int*)(ws + WS_ANCH) + (size_t)b * Kc;
  const float* cb = (const float*)(ws + WS_BOX) + (size_t)b * Kc * 4;
  int anchor = ca[jc];
  float ce = ctr[(size_t)b * Nc + anchor];
  float* ol = out + ((size_t)b * Pc + p) * Cc;
  if (tid < Cc) {
    float l = logits[((size_t)b * Nc + anchor) * Cc + tid];
    ol[tid] = ok ? sqrtf(l * ce) : 0.f;       // full fused row, incl. class 0
  }
  float* ob = out + (size_t)Bc * Pc * Cc + ((size_t)b * Pc + p) * 4;
  if (tid < 4) ob[tid] = ok ? cb[jc * 4 + tid] : 0.f;
}

extern "C" void kernel_launch(void* const* d_in, const int* in_sizes, int n_in,
                              void* d_out, int out_size, void* d_ws, size_t ws_size,
                              hipStream_t stream) {
  const float* logits  = (const float*)d_in[0];   // [B,N,C]
  const float* regress = (const float*)d_in[1];   // [B,N,4]
  const float* points  = (const float*)d_in[2];   // [N,2]
  const float* ctr     = (const float*)d_in[3];   // [B,N,1]
  float* out = (float*)d_out;                     // [B,P,C] ++ [B,P,4]
  uint32_t* ws = (uint32_t*)d_ws;
  unsigned short* mask = (unsigned short*)((char*)d_ws + MASK_BYTE);

  dim3 sgrid((NCc + 2047) / 2048, Bc);
  k_init<<<72, 256, 0, stream>>>(ws);
  k_hist<<<sgrid, 256, 0, stream>>>(logits, ctr, ws, 1);
  k_select<<<Bc, 32, 0, stream>>>(ws, 1);
  k_hist<<<sgrid, 256, 0, stream>>>(logits, ctr, ws, 2);
  k_select<<<Bc, 32, 0, stream>>>(ws, 2);
  k_hist<<<sgrid, 256, 0, stream>>>(logits, ctr, ws, 3);
  k_select<<<Bc, 32, 0, stream>>>(ws, 3);
  k_compact<<<sgrid, 256, 0, stream>>>(logits, ctr, ws);
  k_sort_decode<<<Bc, 1024, 0, stream>>>(regress, points, ws);
  k_iou_mask<<<dim3(8192, Bc), 256, 0, stream>>>(ws, mask);
  k_nms<<<Bc, 128, 0, stream>>>(ws, mask);
  k_gather<<<dim3(Pc, Bc), 128, 0, stream>>>(logits, ctr, ws, out);
}